// SpectralConv3d_dse_71949292142794
// MI455X (gfx1250) — compile-verified
//
#include <hip/hip_runtime.h>

typedef float v2f __attribute__((ext_vector_type(2)));
typedef float v4f __attribute__((ext_vector_type(4)));
typedef float v8f __attribute__((ext_vector_type(8)));

// D = A(16x4 f32) * B(4x16 f32) + C, full-fp32 CDNA5 WMMA
#define WMMA_F32(A_, B_, C_) \
  __builtin_amdgcn_wmma_f32_16x16x4_f32(false, (A_), false, (B_), (short)0, (C_), false, false)

// Hardware trig: v_sin_f32/v_cos_f32 evaluate sin/cos of (2*pi * input),
// i.e. input in revolutions. Phases t = K.pos are already revolutions.
__device__ __forceinline__ float hw_sin(float t) { return __builtin_amdgcn_sinf(t); }
__device__ __forceinline__ float hw_cos(float t) { return __builtin_amdgcn_cosf(t); }

constexpr int kB  = 2;
constexpr int kC  = 32;
constexpr int kN  = 8192;
constexpr int kMH = 864;   // 12*12*6 retained (half-z) modes

// FFT-ordered frequencies for flat mode index m = (x*12 + y)*6 + z
__device__ __forceinline__ void mode_freqs(int m, float& fx, float& fy, float& fz) {
  const int x = m / 72;
  const int y = (m / 6) % 12;
  const int z = m % 6;
  fx = (float)(x < 6 ? x : x - 12);
  fy = (float)(y < 6 ? y : y - 12);
  fz = (float)z;
}

// ---- Prep A: x[b,c,n] -> xT2[b][n/2][c][n&1] (pair-interleaved) so each
// WMMA B-vector {x[na][c], x[na+1][c]} is one aligned b64 load. ----
__global__ void __launch_bounds__(256) k_transpose(const float* __restrict__ x,
                                                   float* __restrict__ xT2) {
  const int t = blockIdx.x * blockDim.x + threadIdx.x;   // e = b*N*C + n*C + c
  const int c = t & 31;
  const int n = (t >> 5) & (kN - 1);
  const int b = t >> 18;                                  // N*C = 2^18
  const float v = x[((size_t)(b * kC + c) << 13) + n];
  xT2[((size_t)b << 18) + (size_t)(n >> 1) * 64 + c * 2 + (n & 1)] = v;
}

// ---- Prep B: pos[b,n,3] (AoS) -> posT[b,3,n] (SoA) for b64 pair loads ----
__global__ void __launch_bounds__(256) k_pos_soa(const float* __restrict__ pos,
                                                 float* __restrict__ posT) {
  const int t = blockIdx.x * blockDim.x + threadIdx.x;   // b*N + n, < 16384
  const int n = t & (kN - 1);
  const int b = t >> 13;
  const float* p = pos + (size_t)t * 3;
  float* o = posT + (size_t)b * 3 * kN + n;
  o[0 * kN] = p[0];
  o[1 * kN] = p[1];
  o[2 * kN] = p[2];
}

// ---- Prep C: per-mode frequency table {fx, fy, fz, 0} ----
__global__ void __launch_bounds__(256) k_modetab(float* __restrict__ fk) {
  const int m = blockIdx.x * blockDim.x + threadIdx.x;
  if (m >= kMH) return;
  float fx, fy, fz;
  mode_freqs(m, fx, fy, fz);
  float* o = fk + (size_t)m * 4;
  o[0] = fx; o[1] = fy; o[2] = fz; o[3] = 0.f;
}

// ---- Stage 1: forward NUFFT partials via V_WMMA_F32_16X16X4_F32 ----
// grid (54 mode-tiles, B, 2), block 256 (8 waves). Wave = one K-slab of 512 samples.
// A[mode_row, k=sample] = cos/sin(2*pi*K_m . pos_n), B[k=sample, c] = x[b,c,n].
// All hot-loop addresses are pointer bumps with constant strides.
__global__ void __launch_bounds__(256) k_forward(const float* __restrict__ posT,
                                                 const float* __restrict__ xT2,
                                                 float* __restrict__ part_re,
                                                 float* __restrict__ part_sn) {
  const int lane = threadIdx.x & 31;
  const int wave = threadIdx.x >> 5;
  const int mt   = blockIdx.x;                 // mode tile 0..53
  const int b    = blockIdx.y;
  const int s    = blockIdx.z * 8 + wave;      // K-slab 0..15
  const int half = lane >> 4;                  // A/B upper-half lanes hold K=2,3
  const int c0   = lane & 15;
  const int mode = mt * 16 + c0;               // this lane's A row (mode)
  float fkx, fky, fkz;
  mode_freqs(mode, fkx, fky, fkz);

  const int n_begin = s * 512;
  const float* px_p = posT + (size_t)b * 3 * kN + n_begin + half * 2;
  const float* py_p = px_p + kN;
  const float* pz_p = py_p + kN;
  // pair-interleaved B base for this lane: row (n_begin/2 + half), column c0
  const float* bb = xT2 + ((size_t)b << 18) +
                    (size_t)(n_begin >> 1) * 64 + half * 64 + c0 * 2;

  v8f aRe0 = {}, aSn0 = {}, aRe1 = {}, aSn1 = {};
  for (int step = 0; step < 128; ++step) {
    const v2f pxv = *(const v2f*)px_p;
    const v2f pyv = *(const v2f*)py_p;
    const v2f pzv = *(const v2f*)pz_p;
    const float ta = fkx * pxv.x + fky * pyv.x + fkz * pzv.x;  // revolutions
    const float tb = fkx * pxv.y + fky * pyv.y + fkz * pzv.y;
    const v2f Ac = {hw_cos(ta), hw_cos(tb)};
    const v2f As = {hw_sin(ta), hw_sin(tb)};
    const v2f B0 = *(const v2f*)bb;            // {x[na][c0],    x[na+1][c0]}
    const v2f B1 = *(const v2f*)(bb + 32);     // {x[na][c0+16], x[na+1][c0+16]}
    aRe0 = WMMA_F32(Ac, B0, aRe0);   // Re += x * cos
    aSn0 = WMMA_F32(As, B0, aSn0);   // Sn += x * sin  (x_ft.im = -Sn)
    aRe1 = WMMA_F32(Ac, B1, aRe1);
    aSn1 = WMMA_F32(As, B1, aSn1);
    px_p += 4; py_p += 4; pz_p += 4;
    bb += 128;
  }

  // D layout: VGPR r, lanes 0-15 -> row r, lanes 16-31 -> row r+8; col = lane&15
  float* pr = part_re + (size_t)(s * 2 + b) * (kMH * kC);
  float* ps = part_sn + (size_t)(s * 2 + b) * (kMH * kC);
  const int mrow = mt * 16 + half * 8;
#pragma unroll
  for (int r = 0; r < 8; ++r) {
    const int i0 = (mrow + r) * 32 + c0;
    pr[i0]      = aRe0[r];
    ps[i0]      = aSn0[r];
    pr[i0 + 16] = aRe1[r];
    ps[i0 + 16] = aSn1[r];
  }
}

// ---- Stage 2: reduce 16 K-slabs ----
__global__ void __launch_bounds__(256) k_reduce(const float* __restrict__ part_re,
                                                const float* __restrict__ part_sn,
                                                float* __restrict__ ft_re,
                                                float* __restrict__ ft_sn) {
  const int e = blockIdx.x * blockDim.x + threadIdx.x;   // < 2*864*32 = 55296
  float r = 0.f, s = 0.f;
#pragma unroll
  for (int k = 0; k < 16; ++k) {
    r += part_re[(size_t)k * 55296 + e];
    s += part_sn[(size_t)k * 55296 + e];
  }
  ft_re[e] = r;
  ft_sn[e] = s;
}

// ---- Stage 3: per-mode 32x32 complex weight multiply (quadrant select) ----
// One wave per (mode, b); lane = output channel. Writes Re and NEGATED Im
// in pair-interleaved layout xf2[b][m/2][c][m&1] for b64 B-loads in stage 4.
__global__ void __launch_bounds__(32) k_specmul(const float* __restrict__ ft_re,
                                                const float* __restrict__ ft_sn,
                                                const float* __restrict__ w1,
                                                const float* __restrict__ w2,
                                                const float* __restrict__ w3,
                                                const float* __restrict__ w4,
                                                float* __restrict__ xf2_re,
                                                float* __restrict__ xf2_imn) {
  const int m = blockIdx.x, b = blockIdx.y, o = threadIdx.x;
  const int x = m / 72, y = (m / 6) % 12, z = m % 6;
  const float* w = (x < 6) ? ((y < 6) ? w1 : w3) : ((y < 6) ? w2 : w4);
  const int widx = (((x < 6 ? x : x - 6) * 6 + (y < 6 ? y : y - 6)) * 6 + z) * 2;
  const size_t fbase = (size_t)(b * kMH + m) * kC;
  float accRe = 0.f, accIm = 0.f;
#pragma unroll 4
  for (int i = 0; i < 32; ++i) {
    const float are = ft_re[fbase + i];
    const float asn = ft_sn[fbase + i];            // x_ft.im = -asn
    const float* wp = w + (size_t)(i * 32 + o) * 432 + widx;  // 216 cplx * 2
    const float wre = wp[0], wim = wp[1];
    accRe += are * wre + asn * wim;                // are*wre - (-asn)*wim
    accIm += are * wim - asn * wre;                // are*wim + (-asn)*wre
  }
  const size_t oidx = ((size_t)b << 15) + (size_t)(m >> 1) * 64 + o * 2 + (m & 1);
  xf2_re[oidx]  = accRe;                           // batch stride 2^15; per batch (864/2)*32*2 = 27648
  xf2_imn[oidx] = -accIm;
}

// ---- Stage 4: inverse NUFFT. y[b,c,n] = (ReS[c] + ReS[31-c]) * 2/N,
// ReS = sum_m cos(th)*xf_re + sin(th)*(-xf_im), th = +2*pi*K_m.pos_n.
// grid (64, B), block 256 (8 waves); wave = one 16-sample tile, K = 864 modes.
__global__ void __launch_bounds__(256) k_inverse(const float* __restrict__ pos,
                                                 const float* __restrict__ fk,
                                                 const float* __restrict__ xf2_re,
                                                 const float* __restrict__ xf2_imn,
                                                 float* __restrict__ y) {
  const int lane = threadIdx.x & 31;
  const int wave = threadIdx.x >> 5;
  const int b    = blockIdx.y;
  const int n0   = (blockIdx.x * 8 + wave) * 16;
  const int half = lane >> 4;
  const int c0   = lane & 15;

  const float* pp = pos + ((size_t)b * kN + (n0 + c0)) * 3;  // this lane's sample row
  const float px = pp[0], py = pp[1], pz = pp[2];

  // pointer-bump bases: fk advances 16 floats/step, xf rows 128 floats/step
  const float* fk_p = fk + half * 8;
  const float* rb = xf2_re  + ((size_t)b << 15) + half * 64 + c0 * 2;
  const float* ib = xf2_imn + ((size_t)b << 15) + half * 64 + c0 * 2;

  v8f acc0 = {}, acc1 = {};
  for (int ms = 0; ms < 216; ++ms) {
    const v4f fa = *(const v4f*)fk_p;             // broadcast, L0-hot
    const v4f fb = *(const v4f*)(fk_p + 4);
    const float ta = fa.x * px + fa.y * py + fa.z * pz;  // revolutions
    const float tb = fb.x * px + fb.y * py + fb.z * pz;
    const v2f Ac = {hw_cos(ta), hw_cos(tb)};
    const v2f As = {hw_sin(ta), hw_sin(tb)};
    const v2f Br0 = *(const v2f*)rb;              // {re[mA][c0],    re[mB][c0]}
    const v2f Br1 = *(const v2f*)(rb + 32);       // {re[mA][c0+16], re[mB][c0+16]}
    const v2f Bi0 = *(const v2f*)ib;
    const v2f Bi1 = *(const v2f*)(ib + 32);
    acc0 = WMMA_F32(Ac, Br0, acc0);
    acc0 = WMMA_F32(As, Bi0, acc0);
    acc1 = WMMA_F32(Ac, Br1, acc1);
    acc1 = WMMA_F32(As, Bi1, acc1);
    fk_p += 16;
    rb += 128;
    ib += 128;
  }

  const float scale = 2.0f / (float)kN;
  float* yb = y + ((size_t)b << 18);
#pragma unroll
  for (int r = 0; r < 8; ++r) {
    const int n = n0 + r + half * 8;
    // Hermitian combine: channel c <-> 31-c lives at lane^15 (same row half)
    const float p0 = __shfl_xor(acc1[r], 15, 32);
    const float p1 = __shfl_xor(acc0[r], 15, 32);
    yb[(size_t)c0 * kN + n]        = (acc0[r] + p0) * scale;
    yb[(size_t)(c0 + 16) * kN + n] = (acc1[r] + p1) * scale;
  }
}

extern "C" void kernel_launch(void* const* d_in, const int* in_sizes, int n_in,
                              void* d_out, int out_size, void* d_ws, size_t ws_size,
                              hipStream_t stream) {
  (void)in_sizes; (void)n_in; (void)out_size; (void)ws_size;
  const float* x   = (const float*)d_in[0];
  const float* pos = (const float*)d_in[1];
  const float* w1  = (const float*)d_in[2];
  const float* w2  = (const float*)d_in[3];
  const float* w3  = (const float*)d_in[4];
  const float* w4  = (const float*)d_in[5];
  float* y = (float*)d_out;

  float* W       = (float*)d_ws;
  float* xT2     = W;                   // 524288 floats (2 MB), pair-interleaved
  float* part_re = xT2 + 524288;        // 16*2*864*32 = 884736
  float* part_sn = part_re + 884736;    // 884736
  float* ft_re   = part_sn + 884736;    // 55296
  float* ft_sn   = ft_re + 55296;       // 55296
  float* xf2_re  = ft_sn + 55296;       // 55296, pair-interleaved
  float* xf2_imn = xf2_re + 55296;      // 55296, pair-interleaved
  float* posT    = xf2_imn + 55296;     // 2*3*8192 = 49152
  float* fktab   = posT + 49152;        // 864*4 = 3456   (total ~10.3 MB, L2-resident)

  k_transpose<<<2048, 256, 0, stream>>>(x, xT2);
  k_pos_soa<<<64, 256, 0, stream>>>(pos, posT);
  k_modetab<<<4, 256, 0, stream>>>(fktab);
  k_forward<<<dim3(54, 2, 2), 256, 0, stream>>>(posT, xT2, part_re, part_sn);
  k_reduce<<<216, 256, 0, stream>>>(part_re, part_sn, ft_re, ft_sn);
  k_specmul<<<dim3(864, 2), 32, 0, stream>>>(ft_re, ft_sn, w1, w2, w3, w4, xf2_re, xf2_imn);
  k_inverse<<<dim3(64, 2), 256, 0, stream>>>(pos, fktab, xf2_re, xf2_imn, y);
}